// Model_44805098832275
// MI455X (gfx1250) — compile-verified
//
#include <hip/hip_runtime.h>

// ---------------------------------------------------------------------------
// ST-GCN forward for MI455X (gfx1250, wave32, WMMA).
// Matrix math: V_WMMA_F32_16X16X4_F32 (full f32 fidelity; model is memory-
// bound at 23.3 TB/s so the f32 tensor pipe is nowhere near the ceiling).
//
// d_in leaf order (JAX pytree, dict keys sorted alphabetically):
//   [0] x [32,3,512,25]   [1] A [3,25,25]
//   blocks 0..9: W, b, bn.beta, bn.gamma, bn.mean, bn.var,
//                (+ res.W, res.b, res.bn.beta/gamma/mean/var for conv blocks 4,7)
//   data_bn: beta, gamma, mean, var    logits: W, b
//
// d_ws layout (floats): XA[M*256] | XB[M*256] | Hc[CHUNK_ROWS*768] | Rw[M*256]
//   => requires ws_size >= ~1.42 GB.
// ---------------------------------------------------------------------------

#define EPSBN 1e-3f
#define NB   32
#define CIN  3
#define TT   512
#define VV   25
#define MTOT (NB*TT*VV)         // 409600 rows
#define NTG  (NB*TT)            // 16384 (n,t) groups
#define CHUNK_G 2048            // (n,t) groups per chunk
#define CHUNK_ROWS (CHUNK_G*VV) // 51200 rows -> H chunk = 157MB (L2-resident)
#define LDSPITCH 18             // row pitch in floats: conflict-free, 8B aligned

typedef float v2f __attribute__((ext_vector_type(2)));
typedef float v8f __attribute__((ext_vector_type(8)));

__device__ __forceinline__ v8f wmma_f32(v2f a, v2f b, v8f c) {
  // D = A(16x4, f32) * B(4x16, f32) + C(16x16, f32)
  return __builtin_amdgcn_wmma_f32_16x16x4_f32(false, a, false, b,
                                               (short)0, c, false, false);
}

// --------------------------- data BN + transpose ---------------------------
// x[N,C,T,V] -> X0[(n*T+t)*V+v, 4] with BN over channel (v*3+c); col 3 = 0.
__global__ void __launch_bounds__(256)
k_data_bn(const float* __restrict__ x,
          const float* __restrict__ beta, const float* __restrict__ gamma,
          const float* __restrict__ mean, const float* __restrict__ var,
          float* __restrict__ X0)
{
  int m = blockIdx.x * blockDim.x + threadIdx.x;
  if (m >= MTOT) return;
  int v  = m % VV;
  int nt = m / VV;
  int t  = nt % TT;
  int n  = nt / TT;
  float4 o;
  float vals[3];
#pragma unroll
  for (int c = 0; c < 3; ++c) {
    int ch = v * 3 + c;
    float s  = gamma[ch] * rsqrtf(var[ch] + EPSBN);
    float sh = beta[ch] - mean[ch] * s;
    float xv = x[((size_t)(n * 3 + c) * TT + t) * VV + v];
    vals[c] = xv * s + sh;
  }
  o.x = vals[0]; o.y = vals[1]; o.z = vals[2]; o.w = 0.0f;
  *(float4*)(X0 + (size_t)m * 4) = o;
}

// ------------------------------ WMMA GEMM ----------------------------------
// Out[row, col] = (sum_k X[row,k] * W[k,col] + bias[col]) [* BN scale + shift]
// Block tile 128x128: 8 waves as 4(m) x 2(n); wave tile 32 x 64.
// X tile (128x16) and W tile (16x128, stored transposed) are double-buffered
// in LDS with pitch-18 rows (16 distinct banks for the 16 fragment lanes).
template <bool WITH_BN>
__global__ void __launch_bounds__(256)
k_gemm(const float* __restrict__ X, int lda,
       const float* __restrict__ W, int Kdim, int Ncols,
       const float* __restrict__ bias,
       const float* __restrict__ bng, const float* __restrict__ bnb,
       const float* __restrict__ bnm, const float* __restrict__ bnv,
       float* __restrict__ Out, int ldo)
{
  __shared__ float Xs[2][128 * LDSPITCH];
  __shared__ float Wt[2][128 * LDSPITCH];

  const int tid  = threadIdx.x;
  const int lane = tid & 31;
  const int wave = tid >> 5;
  const int mw = wave & 3;
  const int nw = wave >> 2;
  const int lo = lane & 15;
  const int hi = lane >> 4;           // 0/1: K offset +0 / +2 in fragments
  const int Kpad = (Kdim + 3) & ~3;   // X buffer is stride-padded with zeros
  const int nchunks = (Kpad + 15) >> 4;
  const int rowg0 = blockIdx.x * 128;
  const int colg0 = blockIdx.y * 128;

  // Cooperative fill of K-chunk `c` (16 K values, zero-padded) into buffer b.
  auto fill = [&](int b, int c) {
    const int kc0 = c << 4;
#pragma unroll
    for (int it = 0; it < 8; ++it) {            // X tile: 128 rows x 16 k
      int idx = tid + it * 256;                 // 0..2047
      int row = idx >> 4, kk = idx & 15;
      int k = kc0 + kk;
      float v = 0.0f;
      if (k < Kpad) v = X[(size_t)(rowg0 + row) * lda + k];
      Xs[b][row * LDSPITCH + kk] = v;
    }
#pragma unroll
    for (int it = 0; it < 8; ++it) {            // W tile transposed: 128c x 16k
      int idx = tid + it * 256;
      int col = idx & 127, kk = idx >> 7;
      int k = kc0 + kk, cg = colg0 + col;
      float v = 0.0f;
      if (k < Kdim && cg < Ncols) v = W[(size_t)k * Ncols + cg];
      Wt[b][col * LDSPITCH + kk] = v;
    }
    // Prefetch the K-chunk after next of X (speculative, dropped on fault).
    if (kc0 + 32 < Kpad)
      __builtin_prefetch(&X[(size_t)(rowg0 + (tid >> 1)) * lda + kc0 + 32], 0, 1);
  };

  v8f acc[2][4];
#pragma unroll
  for (int i = 0; i < 2; ++i)
#pragma unroll
    for (int j = 0; j < 4; ++j)
#pragma unroll
      for (int e = 0; e < 8; ++e) acc[i][j][e] = 0.0f;

  fill(0, 0);
  __syncthreads();

  for (int c = 0; c < nchunks; ++c) {
    if (c + 1 < nchunks) fill((c + 1) & 1, c + 1);
    const float* xb = Xs[c & 1];
    const float* wb = Wt[c & 1];
#pragma unroll
    for (int kk = 0; kk < 16; kk += 4) {        // zero pad => always 4 steps
      const int ka = kk + (hi << 1);
      v2f a[2], b[4];
#pragma unroll
      for (int mt = 0; mt < 2; ++mt)            // A: M = lane%16, K striped
        a[mt] = *(const v2f*)&xb[(mw * 32 + mt * 16 + lo) * LDSPITCH + ka];
#pragma unroll
      for (int nt = 0; nt < 4; ++nt)            // B: N = lane%16, K striped
        b[nt] = *(const v2f*)&wb[(nw * 64 + nt * 16 + lo) * LDSPITCH + ka];
#pragma unroll
      for (int mt = 0; mt < 2; ++mt)
#pragma unroll
        for (int nt = 0; nt < 4; ++nt)
          acc[mt][nt] = wmma_f32(a[mt], b[nt], acc[mt][nt]);
    }
    __syncthreads();
  }

#pragma unroll
  for (int nt = 0; nt < 4; ++nt) {
    int col = colg0 + nw * 64 + nt * 16 + lo;
    if (col >= Ncols) continue;                 // tiles are 16-aligned: uniform
    float sc = 1.0f;
    float sh = bias[col];
    if (WITH_BN) {                              // fold BN: (acc + b)*s + t
      float s = bng[col] * rsqrtf(bnv[col] + EPSBN);
      sh = sh * s + (bnb[col] - bnm[col] * s);
      sc = s;
    }
#pragma unroll
    for (int mt = 0; mt < 2; ++mt) {
      int rbase = rowg0 + mw * 32 + mt * 16 + (hi << 3);  // row = r + 8*hi
#pragma unroll
      for (int r = 0; r < 8; ++r) {
        int row = rbase + r;
        Out[(size_t)row * ldo + col] = acc[mt][nt][r] * sc + sh;
      }
    }
  }
}

// ------------------- graph conv + BN + residual + ReLU ---------------------
// Per (n,t) group: Out[w, c] = relu(BN(sum_{kv} AT[w,kv] * Hh[kv,c]) + res)
// AT (25x75 zero-padded to 32x76) staged in LDS; Hh re-indexes the H slab.
// MODE: 0 = zero residual, 1 = identity, 2 = projected (Res buffer).
template <int MODE>
__global__ void __launch_bounds__(128)
k_graphconv(const float* __restrict__ H, int Kf, int f,
            const float* __restrict__ A,
            const float* __restrict__ bng, const float* __restrict__ bnb,
            const float* __restrict__ bnm, const float* __restrict__ bnv,
            const float* __restrict__ Res, int res_ld,
            float* __restrict__ Out, int group0)
{
  __shared__ float ATs[32][76];
  for (int idx = threadIdx.x; idx < 32 * 76; idx += 128) {
    int w = idx / 76, kv = idx - w * 76;
    float val = 0.0f;
    if (w < 25 && kv < 75) {
      int k = kv / 25, v = kv - k * 25;
      val = A[k * 625 + v * 25 + w];   // AT[w][k*25+v] = A[k,v,w]
    }
    ATs[w][kv] = val;
  }
  __syncthreads();

  const int lane = threadIdx.x & 31;
  const int wave = threadIdx.x >> 5;
  const int lo = lane & 15, hi = lane >> 4;
  const int c  = blockIdx.y * 64 + wave * 16 + lo;
  const int gl = blockIdx.x;                       // chunk-local group
  const size_t gg = (size_t)group0 + gl;           // global group
  const float* Hg = H + (size_t)gl * 25 * Kf;

  v8f acc[2];
#pragma unroll
  for (int mt = 0; mt < 2; ++mt)
#pragma unroll
    for (int e = 0; e < 8; ++e) acc[mt][e] = 0.0f;

  for (int s = 0; s < 19; ++s) {                   // K = 76 (75 + 1 zero pad)
    int kvb = s * 4 + (hi << 1);
    v2f a[2];
#pragma unroll
    for (int mt = 0; mt < 2; ++mt)
      a[mt] = *(const v2f*)&ATs[mt * 16 + lo][kvb];
    v2f b;
#pragma unroll
    for (int j = 0; j < 2; ++j) {
      int kv = kvb + j;
      float hv = 0.0f;
      if (kv < 75) {
        int k = kv / 25, v = kv - k * 25;
        hv = Hg[(size_t)v * Kf + k * f + c];       // Hh[kv, c]
      }
      b[j] = hv;
    }
#pragma unroll
    for (int mt = 0; mt < 2; ++mt)
      acc[mt] = wmma_f32(a[mt], b, acc[mt]);
  }

  float s = bng[c] * rsqrtf(bnv[c] + EPSBN);
  float t = bnb[c] - bnm[c] * s;
#pragma unroll
  for (int mt = 0; mt < 2; ++mt) {
#pragma unroll
    for (int r = 0; r < 8; ++r) {
      int w = mt * 16 + (hi << 3) + r;
      if (w < 25) {
        float val = acc[mt][r] * s + t;
        size_t orow = gg * 25 + w;
        if (MODE == 1) val += Res[orow * res_ld + c];
        if (MODE == 2) val += Res[orow * f + c];
        Out[orow * f + c] = fmaxf(val, 0.0f);
      }
    }
  }
}

// ----------------------- global mean pool + logits -------------------------
__global__ void __launch_bounds__(256)
k_pool_logits(const float* __restrict__ Xf,
              const float* __restrict__ Wl, const float* __restrict__ bl,
              float* __restrict__ out)
{
  __shared__ float gsh[256];
  int n = blockIdx.x;
  int c = threadIdx.x;
  const float* base = Xf + (size_t)n * 12800 * 256 + c;
  float a0 = 0, a1 = 0, a2 = 0, a3 = 0;
  for (int r = 0; r < 12800; r += 4) {
    a0 += base[(size_t)(r + 0) * 256];
    a1 += base[(size_t)(r + 1) * 256];
    a2 += base[(size_t)(r + 2) * 256];
    a3 += base[(size_t)(r + 3) * 256];
  }
  gsh[c] = (a0 + a1 + a2 + a3) * (1.0f / 12800.0f);
  __syncthreads();
  if (c < 10) {
    float o = bl[c];
    for (int j = 0; j < 256; ++j) o += gsh[j] * Wl[j * 10 + c];
    out[n * 10 + c] = o;
  }
}

// ------------------------------- launcher ----------------------------------
extern "C" void kernel_launch(void* const* d_in, const int* in_sizes, int n_in,
                              void* d_out, int out_size, void* d_ws, size_t ws_size,
                              hipStream_t stream)
{
  (void)in_sizes; (void)n_in; (void)out_size; (void)ws_size;
  const float* x = (const float*)d_in[0];
  const float* A = (const float*)d_in[1];

  struct Cfg { int cin, f, mode; };
  const Cfg cfg[10] = {{3,64,0},{64,64,1},{64,64,1},{64,64,1},{64,128,2},
                       {128,128,1},{128,128,1},{128,256,2},{256,256,1},{256,256,1}};

  const float *BW[10], *Bb[10], *Bbeta[10], *Bgam[10], *Bmu[10], *Bvar[10];
  const float *RW[10], *Rbv[10], *Rbeta[10], *Rgam[10], *Rmu[10], *Rvar[10];
  int pi = 2;
  for (int i = 0; i < 10; ++i) {
    BW[i]    = (const float*)d_in[pi++];
    Bb[i]    = (const float*)d_in[pi++];
    Bbeta[i] = (const float*)d_in[pi++];
    Bgam[i]  = (const float*)d_in[pi++];
    Bmu[i]   = (const float*)d_in[pi++];
    Bvar[i]  = (const float*)d_in[pi++];
    if (cfg[i].mode == 2) {
      RW[i]    = (const float*)d_in[pi++];
      Rbv[i]   = (const float*)d_in[pi++];
      Rbeta[i] = (const float*)d_in[pi++];
      Rgam[i]  = (const float*)d_in[pi++];
      Rmu[i]   = (const float*)d_in[pi++];
      Rvar[i]  = (const float*)d_in[pi++];
    } else {
      RW[i] = Rbv[i] = Rbeta[i] = Rgam[i] = Rmu[i] = Rvar[i] = nullptr;
    }
  }
  const float* db_beta = (const float*)d_in[pi++];
  const float* db_gam  = (const float*)d_in[pi++];
  const float* db_mu   = (const float*)d_in[pi++];
  const float* db_var  = (const float*)d_in[pi++];
  const float* LW      = (const float*)d_in[pi++];
  const float* Lb      = (const float*)d_in[pi++];

  float* XA = (float*)d_ws;
  float* XB = XA + (size_t)MTOT * 256;
  float* Hc = XB + (size_t)MTOT * 256;
  float* Rw = Hc + (size_t)CHUNK_ROWS * 768;

  k_data_bn<<<MTOT / 256, 256, 0, stream>>>(x, db_beta, db_gam, db_mu, db_var, XA);

  for (int i = 0; i < 10; ++i) {
    const float* Xcur = (i & 1) ? XB : XA;
    float*       Xnxt = (i & 1) ? XA : XB;
    int lda  = (i == 0) ? 4 : cfg[i].cin;
    int f    = cfg[i].f;
    int cin  = cfg[i].cin;
    int Kf   = 3 * f;
    int mode = cfg[i].mode;

    if (mode == 2) {   // residual 1x1 conv + BN projection into Rw
      dim3 gr(MTOT / 128, (f + 127) / 128);
      k_gemm<true><<<gr, 256, 0, stream>>>(Xcur, lda, RW[i], cin, f, Rbv[i],
                                           Rgam[i], Rbeta[i], Rmu[i], Rvar[i],
                                           Rw, f);
    }
    for (int ch = 0; ch < NTG / CHUNK_G; ++ch) {
      const float* Xchunk = Xcur + (size_t)ch * CHUNK_ROWS * lda;
      dim3 g1(CHUNK_ROWS / 128, (Kf + 127) / 128);
      k_gemm<false><<<g1, 256, 0, stream>>>(Xchunk, lda, BW[i], cin, Kf, Bb[i],
                                            nullptr, nullptr, nullptr, nullptr,
                                            Hc, Kf);
      dim3 g2(CHUNK_G, f / 64);
      if (mode == 0) {
        k_graphconv<0><<<g2, 128, 0, stream>>>(Hc, Kf, f, A,
                                               Bgam[i], Bbeta[i], Bmu[i], Bvar[i],
                                               nullptr, 0, Xnxt, ch * CHUNK_G);
      } else if (mode == 1) {
        k_graphconv<1><<<g2, 128, 0, stream>>>(Hc, Kf, f, A,
                                               Bgam[i], Bbeta[i], Bmu[i], Bvar[i],
                                               Xcur, cin, Xnxt, ch * CHUNK_G);
      } else {
        k_graphconv<2><<<g2, 128, 0, stream>>>(Hc, Kf, f, A,
                                               Bgam[i], Bbeta[i], Bmu[i], Bvar[i],
                                               Rw, f, Xnxt, ch * CHUNK_G);
      }
    }
  }
  // block 9 is odd -> final activations live in XA
  k_pool_logits<<<32, 256, 0, stream>>>(XA, LW, Lb, (float*)d_out);
}